// ChamferLoss_51470888075275
// MI455X (gfx1250) — compile-verified
//
#include <hip/hip_runtime.h>
#include <hip/hip_bf16.h>
#include <math.h>

typedef __attribute__((ext_vector_type(2))) float v2f;
typedef __attribute__((ext_vector_type(8))) float v8f;

// One wave per block. Wave owns TWO 16-row tiles of A (32 points) and streams
// all of B in 16-column tiles. Each B tile costs ONE b96 load + prep and feeds
// TWO v_wmma_f32_16x16x4_f32, each directly producing a 16x16 tile of squared
// distances:
//   A' = [-2ax, -2ay, -2az, 1]  (16x4, K padded)
//   B' = [ bx,   by,   bz, ||b||^2 ]  (4x16)
//   C  = ||a_i||^2 broadcast per row
//   D  = C + A'B' = ||a||^2 - 2 a.b + ||b||^2 = d2(i,j)
//
// The reference's clamp max(d2,0) is monotone, so it commutes with min and is
// applied ONCE to the running mins after the j-loop (keeps the loop lean).
//
// A-matrix 16x4 f32 layout (ISA 7.12.2): lanes 0-15 hold (K0,K1) of row=lane,
// lanes 16-31 hold (K2,K3) of row=lane-16. B 4x16 mirrored. C/D 16x16 f32
// layout: VGPR v = row v (lanes 0-15) / row v+8 (lanes 16-31).
__global__ __launch_bounds__(32)
void chamfer_dir_kernel(const float* __restrict__ A, const float* __restrict__ Bp,
                        int nA, int nB, float* __restrict__ partial)
{
    const int lane = threadIdx.x & 31;
    const int tile = blockIdx.x;     // which 32-row block of A
    const int b    = blockIdx.y;     // batch
    const int i0   = tile * 32;

    const float* __restrict__ Ab = A  + (size_t)b * (size_t)nA * 3u;
    const float* __restrict__ Bb = Bp + (size_t)b * (size_t)nB * 3u;

    const int  p  = lane & 15;       // point index within a 16-tile
    const bool hi = lane >= 16;      // K-half selector

    // ---- per-wave setup: load own A points (two tiles), build A' and C ----
    const float* ap0 = Ab + (size_t)(i0 + p) * 3u;
    const float* ap1 = Ab + (size_t)(i0 + 16 + p) * 3u;
    float a0x = ap0[0], a0y = ap0[1], a0z = ap0[2];
    float a1x = ap1[0], a1y = ap1[1], a1z = ap1[2];
    float anorm0 = a0x * a0x + a0y * a0y + a0z * a0z;
    float anorm1 = a1x * a1x + a1y * a1y + a1z * a1z;

    v2f av0, av1;
    av0.x = hi ? (-2.0f * a0z) : (-2.0f * a0x);   // K2 : K0
    av0.y = hi ? 1.0f          : (-2.0f * a0y);   // K3 : K1
    av1.x = hi ? (-2.0f * a1z) : (-2.0f * a1x);
    av1.y = hi ? 1.0f          : (-2.0f * a1y);

    v8f cinit0, cinit1;
#pragma unroll
    for (int v = 0; v < 8; ++v) {
        // row of c[v] for this lane-half is (v) or (v+8); its norm lives in
        // lane (v) / lane (v+8) (both halves computed it identically).
        const int src = hi ? (v + 8) : v;
        cinit0[v] = __shfl(anorm0, src, 32);
        cinit1[v] = __shfl(anorm1, src, 32);
    }

    float rmin0[8], rmin1[8];
#pragma unroll
    for (int v = 0; v < 8; ++v) {
        rmin0[v] = 3.402823466e38f;
        rmin1[v] = 3.402823466e38f;
    }

    // ---- stream B in 16-point tiles: 1 load, 2 WMMAs per tile ----
#pragma unroll 2
    for (int j0 = 0; j0 < nB; j0 += 16) {
        const float* bp = Bb + (size_t)(j0 + p) * 3u;
        float bx = bp[0], by = bp[1], bz = bp[2];
        float bnorm = bx * bx + by * by + bz * bz;

        v2f bv;
        bv.x = hi ? bz    : bx;   // K2 : K0
        bv.y = hi ? bnorm : by;   // K3 : K1

        v8f d0 = __builtin_amdgcn_wmma_f32_16x16x4_f32(
            false, av0, false, bv, (short)0, cinit0, false, false);
        v8f d1 = __builtin_amdgcn_wmma_f32_16x16x4_f32(
            false, av1, false, bv, (short)0, cinit1, false, false);

#pragma unroll
        for (int v = 0; v < 8; ++v) {
            rmin0[v] = fminf(rmin0[v], d0[v]);
            rmin1[v] = fminf(rmin1[v], d1[v]);
        }
    }

    // ---- clamp once (commutes with min), then finish min over columns ----
#pragma unroll
    for (int v = 0; v < 8; ++v) {
        float m0 = fmaxf(rmin0[v], 0.0f);
        float m1 = fmaxf(rmin1[v], 0.0f);
        m0 = fminf(m0, __shfl_xor(m0, 1, 32));
        m1 = fminf(m1, __shfl_xor(m1, 1, 32));
        m0 = fminf(m0, __shfl_xor(m0, 2, 32));
        m1 = fminf(m1, __shfl_xor(m1, 2, 32));
        m0 = fminf(m0, __shfl_xor(m0, 4, 32));
        m1 = fminf(m1, __shfl_xor(m1, 4, 32));
        m0 = fminf(m0, __shfl_xor(m0, 8, 32));
        m1 = fminf(m1, __shfl_xor(m1, 8, 32));
        rmin0[v] = m0;   // row-min, uniform per 16-lane half
        rmin1[v] = m1;
    }
    // sum the rows held by this half (tile0 rows + tile1 rows), then add the
    // other half's rows.
    float s = (((rmin0[0] + rmin0[1]) + (rmin0[2] + rmin0[3])) +
               ((rmin0[4] + rmin0[5]) + (rmin0[6] + rmin0[7]))) +
              (((rmin1[0] + rmin1[1]) + (rmin1[2] + rmin1[3])) +
               ((rmin1[4] + rmin1[5]) + (rmin1[6] + rmin1[7])));
    s += __shfl_xor(s, 16, 32);

    if (lane == 0)
        partial[(size_t)b * gridDim.x + tile] = s;  // sum_i min_j d2, 32 rows
}

// Deterministic single-block tree reduction of all per-wave partials.
__global__ __launch_bounds__(256)
void chamfer_finalize(const float* __restrict__ partial, int tiles1, int tiles2,
                      int nBatch, float inv1, float inv2, float* __restrict__ out)
{
    __shared__ float red[256];
    const int t = threadIdx.x;
    float acc = 0.0f;

    const int total1 = nBatch * tiles1;
    for (int idx = t; idx < total1; idx += 256)
        acc += partial[idx] * inv1;

    const int total2 = nBatch * tiles2;
    const float* p2 = partial + total1;
    for (int idx = t; idx < total2; idx += 256)
        acc += p2[idx] * inv2;

    red[t] = acc;
    __syncthreads();
#pragma unroll
    for (int s = 128; s > 0; s >>= 1) {
        if (t < s) red[t] += red[t + s];
        __syncthreads();
    }
    if (t == 0)
        out[0] = sqrtf(0.5f * red[0] / (float)nBatch);
}

extern "C" void kernel_launch(void* const* d_in, const int* in_sizes, int n_in,
                              void* d_out, int out_size, void* d_ws, size_t ws_size,
                              hipStream_t stream) {
    (void)n_in; (void)out_size; (void)ws_size;
    const float* yhat = (const float*)d_in[0];   // [B, N, 3]
    const float* y    = (const float*)d_in[1];   // [B, M, 3]
    float* out        = (float*)d_out;           // scalar
    float* partial    = (float*)d_ws;            // [B*tilesA + B*tilesB] floats

    const int B_ = 4;
    const int D_ = 3;
    const int N_ = in_sizes[0] / (B_ * D_);      // 8192
    const int M_ = in_sizes[1] / (B_ * D_);      // 8192
    const int tilesA = N_ / 32;                  // 32 A-rows per wave
    const int tilesB = M_ / 32;

    // fwd: yhat -> y  (sum_i min_j), scaled by 1/N in finalize
    chamfer_dir_kernel<<<dim3(tilesA, B_), dim3(32), 0, stream>>>(
        yhat, y, N_, M_, partial);
    // bwd: y -> yhat  (sum_j min_i), scaled by 1/M in finalize
    chamfer_dir_kernel<<<dim3(tilesB, B_), dim3(32), 0, stream>>>(
        y, yhat, M_, N_, partial + (size_t)B_ * tilesA);

    chamfer_finalize<<<1, 256, 0, stream>>>(
        partial, tilesA, tilesB, B_, 1.0f / (float)N_, 1.0f / (float)M_, out);
}